// HausdorffDistanceLoss_42571715838126
// MI455X (gfx1250) — compile-verified
//
#include <hip/hip_runtime.h>
#include <hip/hip_bf16.h>
#include <math.h>

// ---------------------------------------------------------------------------
// Hausdorff-style loss on gfx1250 (MI455X).
//
// dist^2(pixel, target) = ||p||^2 + ||t||^2 - 2 p.t  ==  rank-2 GEMM
// -> V_WMMA_F32_16X16X4_F32 with
//    A row (pixel m):  [gy, gx, gy^2+gx^2, 1 ]            (16x4, K=4)
//    B col (target n): [-2ty, -2tx, 1, ty^2+tx^2]         (4x16)
//    D = A*B = dist^2 tile (16 pixels x 16 targets), exact in f32
//    (all terms are integers < 2^24, so no rounding; D >= 0 provably).
//
// Each wave owns 4 pixel tiles (64 pixels) -> 4 WMMAs per B-tile LDS load.
// All floating-point accumulation is via fixed-order tree reductions
// (deterministic); only the int target counter uses atomics.
// ---------------------------------------------------------------------------

#define BQ   4
#define HQ   256
#define WQ   256
#define HWQ  (HQ * WQ)
#define NMAXQ 1024

#define TILES_PER_WAVE 4                       // pixel tiles (of 16) per wave
#define WAVES_PER_BLK  8
#define PIX_PER_BLOCK  (16 * TILES_PER_WAVE * WAVES_PER_BLK)   // 512
#define NBLK_C         (HWQ / PIX_PER_BLOCK)                   // 128
#define NH1_PARTS      (NBLK_C * WAVES_PER_BLK * 2)            // 2048 / sample
#define NBLK_A         (HWQ / 256)                             // 256 / sample

typedef float v2f __attribute__((ext_vector_type(2)));
typedef float v8f __attribute__((ext_vector_type(8)));

__device__ __forceinline__ float sigmoidf_fast(float x) {
    return 1.0f / (1.0f + __expf(-x));
}

// ---------------------------------------------------------------------------
// Workspace layout (d_ws), float4 block first for 16B alignment:
//   float4 tgt[B][NMAX]           : packed targets (-2ty, -2tx, 1, tnorm)
//   float  partsH1[B][NH1_PARTS]  : per-wave-half hd1 partials
//   float  partsA [B][NBLK_A]     : per-block sigmoid-sum partials
//   float  partsH2[B][NBLK_A]     : per-block (1-p)@target partials
//   float  loss[B]
//   int    tcount[B]
// ---------------------------------------------------------------------------

__global__ void hd_init_kernel(int* tcount) {
    int t = threadIdx.x;
    if (t < BQ) tcount[t] = 0;
}

// Pass 1: sigmoid sums + target compaction.  grid = (HW/256, B), block = 256.
__global__ void hd_compact_kernel(const float* __restrict__ logits,
                                  const int* __restrict__ targets,
                                  float4* __restrict__ tgt,
                                  int* __restrict__ tcount,
                                  float* __restrict__ partsA,
                                  float* __restrict__ partsH2) {
    const int b   = blockIdx.y;
    const int i   = blockIdx.x * 256 + threadIdx.x;    // flat pixel index
    const int off = b * HWQ + i;

    const float x = logits[off];
    const float p = sigmoidf_fast(x);
    const bool  is_t = (targets[off] == 1);

    if (is_t) {
        const int pos = atomicAdd(&tcount[b], 1);      // int atomic: value det.
        if (pos < NMAXQ) {
            const float ty = (float)(i >> 8);          // i / W
            const float tx = (float)(i & 255);         // i % W
            float4 v;
            v.x = -2.0f * ty;
            v.y = -2.0f * tx;
            v.z = 1.0f;
            v.w = ty * ty + tx * tx;
            tgt[b * NMAXQ + pos] = v;
        }
    }

    // Deterministic fixed-order block reductions.
    __shared__ float redP[256];
    __shared__ float redT[256];
    redP[threadIdx.x] = p;
    redT[threadIdx.x] = is_t ? (1.0f - p) : 0.0f;
    __syncthreads();
    #pragma unroll
    for (int s = 128; s > 0; s >>= 1) {
        if (threadIdx.x < s) {
            redP[threadIdx.x] += redP[threadIdx.x + s];
            redT[threadIdx.x] += redT[threadIdx.x + s];
        }
        __syncthreads();
    }
    if (threadIdx.x == 0) {
        partsA [b * NBLK_A + blockIdx.x] = redP[0];
        partsH2[b * NBLK_A + blockIdx.x] = redT[0];
    }
}

// Pass 2: pad unused target slots with a far sentinel so partial tiles of 16
// never win the min.  grid = B, block = 256.
__global__ void hd_pad_kernel(float4* __restrict__ tgt,
                              const int* __restrict__ tcount) {
    const int b = blockIdx.x;
    const int k = tcount[b];
    float4 sentinel;
    sentinel.x = 0.0f; sentinel.y = 0.0f; sentinel.z = 1.0f;
    sentinel.w = 3.0e8f;   // dist^2 >= 3e8 >> max real dist^2 (~1.3e5)
    for (int s = threadIdx.x; s < NMAXQ; s += 256) {
        if (s >= k) tgt[b * NMAXQ + s] = sentinel;
    }
}

// Pass 3 (WMMA): each wave owns TILES_PER_WAVE pixel tiles and loops over
// target tiles of 16 staged in LDS; 4 WMMAs per B-tile load.
// grid = (NBLK_C, B), block = 256 (8 wave32).
__global__ void __launch_bounds__(256)
hd_wmma_kernel(const float* __restrict__ logits,
               const float4* __restrict__ tgt,
               const int* __restrict__ tcount,
               float* __restrict__ partsH1) {
    const int b     = blockIdx.y;
    const int tid   = threadIdx.x;
    const int lane  = tid & 31;
    const int wave  = tid >> 5;

    // Stage this sample's packed target list in LDS (16 KB of 320 KB WGP LDS).
    __shared__ float4 stgt[NMAXQ];
    #pragma unroll
    for (int k = 0; k < NMAXQ / 256; ++k)
        stgt[tid + k * 256] = tgt[b * NMAXQ + tid + k * 256];
    __syncthreads();

    const int K      = tcount[b];                      // uniform per block
    int ntiles       = (K + 15) >> 4;
    if (ntiles > NMAXQ / 16) ntiles = NMAXQ / 16;
    if (ntiles < 1)          ntiles = 1;

    // 64 consecutive flat pixels per wave (4 tiles of 16).
    const int wbase   = (blockIdx.x * WAVES_PER_BLK + wave) * (16 * TILES_PER_WAVE);
    const int ml      = lane & 15;
    const int halfsel = lane >> 4;   // 0 -> K0,K1 half; 1 -> K2,K3 half

    // A operands (16x4 f32, 2 VGPRs/lane):
    //   lanes 0-15  : M = lane,      (K0, K1) = (gy, gx)
    //   lanes 16-31 : M = lane - 16, (K2, K3) = (gy^2+gx^2, 1)
    v2f amat[TILES_PER_WAVE];
    #pragma unroll
    for (int t = 0; t < TILES_PER_WAVE; ++t) {
        const int pix  = wbase + t * 16 + ml;
        const float gy = (float)(pix >> 8);
        const float gx = (float)(pix & 255);
        if (halfsel == 0) { amat[t].x = gy;                 amat[t].y = gx;   }
        else              { amat[t].x = gy * gy + gx * gx;  amat[t].y = 1.0f; }
    }

    float mind[TILES_PER_WAVE][8];
    #pragma unroll
    for (int t = 0; t < TILES_PER_WAVE; ++t)
        #pragma unroll
        for (int j = 0; j < 8; ++j) mind[t][j] = 3.4e38f;

    for (int tt = 0; tt < ntiles; ++tt) {
        // B operand (4x16 f32, 2 VGPRs/lane), shared by all 4 WMMAs:
        //   lanes 0-15  : rows K0,K1 = (-2ty, -2tx) of target n = lane
        //   lanes 16-31 : rows K2,K3 = (1, tnorm)    of target n = lane - 16
        const float2* sp = ((const float2*)&stgt[tt * 16 + ml]) + halfsel;
        v2f bmat;
        bmat.x = sp->x;
        bmat.y = sp->y;

        #pragma unroll
        for (int t = 0; t < TILES_PER_WAVE; ++t) {
            v8f c = {0.0f, 0.0f, 0.0f, 0.0f, 0.0f, 0.0f, 0.0f, 0.0f};
            v8f d = __builtin_amdgcn_wmma_f32_16x16x4_f32(
                /*neg_a=*/false, amat[t], /*neg_b=*/false, bmat,
                /*c_mod=*/(short)0, c, /*reuse_a=*/false, /*reuse_b=*/false);
            // dist^2 is exact & non-negative in f32 here (integer arithmetic
            // below 2^24) -> no clamp needed.
            #pragma unroll
            for (int j = 0; j < 8; ++j)
                mind[t][j] = fminf(mind[t][j], d[j]);
        }
    }

    // Butterfly min across the 16 lanes of each half (N dimension).
    #pragma unroll
    for (int m = 1; m <= 8; m <<= 1) {
        #pragma unroll
        for (int t = 0; t < TILES_PER_WAVE; ++t)
            #pragma unroll
            for (int j = 0; j < 8; ++j)
                mind[t][j] = fminf(mind[t][j], __shfl_xor(mind[t][j], m, 32));
    }

    // D layout: VGPR j holds M = j + 8*(lane/16).  After the butterfly,
    // lane 0 has rows M=0..7 of each tile, lane 16 has rows M=8..15.
    if (ml == 0) {
        const int boff = b * HWQ;
        float acc = 0.0f;
        #pragma unroll
        for (int t = 0; t < TILES_PER_WAVE; ++t) {
            const int mbase = wbase + t * 16 + (halfsel << 3);
            #pragma unroll
            for (int j = 0; j < 8; ++j) {
                const float x  = logits[boff + mbase + j];
                const float pr = sigmoidf_fast(x);
                acc += pr * sqrtf(mind[t][j]);
            }
        }
        // Unique slot per (block, wave, half) -> deterministic accumulation.
        const int slot = ((blockIdx.x * WAVES_PER_BLK + wave) << 1) | halfsel;
        partsH1[b * NH1_PARTS + slot] = acc;
    }
}

// Pass 4: deterministic per-sample reduction.  grid = B, block = 256.
__global__ void hd_reduce_kernel(const float* __restrict__ partsH1,
                                 const float* __restrict__ partsA,
                                 const float* __restrict__ partsH2,
                                 const int* __restrict__ tcount,
                                 float* __restrict__ loss) {
    const int b   = blockIdx.x;
    const int tid = threadIdx.x;

    __shared__ float r1[256];
    __shared__ float rA[256];
    __shared__ float r2[256];

    float s1 = 0.0f;
    #pragma unroll
    for (int k = 0; k < NH1_PARTS / 256; ++k)
        s1 += partsH1[b * NH1_PARTS + tid + k * 256];
    r1[tid] = s1;
    rA[tid] = partsA [b * NBLK_A + tid];
    r2[tid] = partsH2[b * NBLK_A + tid];
    __syncthreads();
    #pragma unroll
    for (int s = 128; s > 0; s >>= 1) {
        if (tid < s) {
            r1[tid] += r1[tid + s];
            rA[tid] += rA[tid + s];
            r2[tid] += r2[tid + s];
        }
        __syncthreads();
    }
    if (tid == 0) {
        const float hd1 = r1[0] / rA[0];
        const float hd2 = r2[0] / (float)tcount[b];
        loss[b] = hd1 + hd2;
    }
}

// Pass 5: mean over batch.
__global__ void hd_final_kernel(const float* __restrict__ loss,
                                float* __restrict__ out) {
    if (threadIdx.x == 0 && blockIdx.x == 0) {
        float acc = 0.0f;
        #pragma unroll
        for (int b = 0; b < BQ; ++b) acc += loss[b];
        out[0] = acc / (float)BQ;
    }
}

extern "C" void kernel_launch(void* const* d_in, const int* in_sizes, int n_in,
                              void* d_out, int out_size, void* d_ws, size_t ws_size,
                              hipStream_t stream) {
    (void)in_sizes; (void)n_in; (void)out_size; (void)ws_size;

    const float* logits  = (const float*)d_in[0];
    const int*   targets = (const int*)d_in[1];
    float*       out     = (float*)d_out;

    // Workspace carve-up (float4 block first for 16B alignment).
    float4* tgt     = (float4*)d_ws;
    float*  fp      = (float*)((char*)d_ws + (size_t)BQ * NMAXQ * sizeof(float4));
    float*  partsH1 = fp;                       fp += BQ * NH1_PARTS;
    float*  partsA  = fp;                       fp += BQ * NBLK_A;
    float*  partsH2 = fp;                       fp += BQ * NBLK_A;
    float*  loss    = fp;                       fp += BQ;
    int*    tcount  = (int*)fp;

    hd_init_kernel<<<1, 64, 0, stream>>>(tcount);

    dim3 gridA(NBLK_A, BQ);
    hd_compact_kernel<<<gridA, 256, 0, stream>>>(logits, targets, tgt, tcount,
                                                 partsA, partsH2);

    hd_pad_kernel<<<BQ, 256, 0, stream>>>(tgt, tcount);

    dim3 gridC(NBLK_C, BQ);   // 8 waves/block, 64 pixels/wave
    hd_wmma_kernel<<<gridC, 256, 0, stream>>>(logits, tgt, tcount, partsH1);

    hd_reduce_kernel<<<BQ, 256, 0, stream>>>(partsH1, partsA, partsH2, tcount,
                                             loss);
    hd_final_kernel<<<1, 32, 0, stream>>>(loss, out);
}